// GATConvNet_42262478192815
// MI455X (gfx1250) — compile-verified
//
#include <hip/hip_runtime.h>
#include <hip/hip_bf16.h>

typedef __attribute__((ext_vector_type(2))) float v2f;
typedef __attribute__((ext_vector_type(8))) float v8f;

// ---------------------------------------------------------------------------
// FP32 WMMA GEMM: C[M,N] = A[M,K] @ B[K,N], row-major.
// One block handles a full 16-row panel: NT = N/16 waves, wave w owns n-tile w.
// The 16xK A panel is staged in LDS once (row stride K+4 floats -> the 16
// row-readers hit 16 distinct banks), then each wave runs the K/4-step
// V_WMMA_F32_16X16X4_F32 loop reading A frags from LDS and B frags from
// global (B is small and L2-resident).
// Frag layouts (ISA 7.12.2):
//   A 16x4: lane L -> m=L%16, vgpr j -> k = 2*(L>=16)+j
//   B 4x16: lane L -> n=L%16, vgpr j -> k = 2*(L>=16)+j
//   C/D   : vgpr v -> row v + 8*(L>=16), col = L%16
// Requires M%16==0, N%16==0, K%4==0, blockDim.x == 32*(N/16).
// ---------------------------------------------------------------------------
__global__ void wmma_gemm_f32(const float* __restrict__ A,
                              const float* __restrict__ B,
                              float* __restrict__ C,
                              int M, int K, int N) {
  extern __shared__ float ldsA[];            // 16 * (K+4) floats
  const int KP   = K + 4;
  const int tid  = threadIdx.x;
  const int lane = tid & 31;
  const int wave = tid >> 5;
  const int l16  = lane & 15;
  const int half = lane >> 4;                // 0 or 1
  const int m0 = blockIdx.x * 16;
  const int n0 = wave * 16;

  // Cooperative stage of the 16xK A panel into LDS (coalesced per row).
  for (int r = 0; r < 16; ++r) {
    const float* __restrict__ src = A + (size_t)(m0 + r) * (size_t)K;
    float* __restrict__ dst = ldsA + r * KP;
    for (int c = tid; c < K; c += blockDim.x) dst[c] = src[c];
  }
  __syncthreads();

  const float* __restrict__ Afrag = ldsA + l16 * KP;
  v8f acc = {};
  for (int k0 = 0; k0 < K; k0 += 4) {
    const int kb = k0 + half * 2;
    v2f a, b;
    a.x = Afrag[kb];
    a.y = Afrag[kb + 1];
    b.x = B[(size_t)kb * (size_t)N + (n0 + l16)];
    b.y = B[(size_t)(kb + 1) * (size_t)N + (n0 + l16)];
    acc = __builtin_amdgcn_wmma_f32_16x16x4_f32(
        /*neg_a=*/false, a, /*neg_b=*/false, b,
        /*c_mod=*/(short)0, acc, /*reuse_a=*/false, /*reuse_b=*/false);
  }
  float* __restrict__ Crow = C + (size_t)(m0 + half * 8) * (size_t)N + (n0 + l16);
#pragma unroll
  for (int v = 0; v < 8; ++v)
    Crow[(size_t)v * (size_t)N] = acc[v];
}

// ---------------------------------------------------------------------------
// Fill a float buffer with a constant.
// ---------------------------------------------------------------------------
__global__ void fill_f32(float* __restrict__ p, float v, long long count) {
  long long i = (long long)blockIdx.x * blockDim.x + threadIdx.x;
  if (i < count) p[i] = v;
}

// ---------------------------------------------------------------------------
// Per-(node,head) attention scores: s_src[n,h] = <h[n,h,:], a_src[h,:]>
// H hardcoded to 8 (both layers).
// ---------------------------------------------------------------------------
__global__ void attn_scores(const float* __restrict__ h,
                            const float* __restrict__ a_src,
                            const float* __restrict__ a_dst,
                            float* __restrict__ ssrc,
                            float* __restrict__ sdst,
                            int n, int C) {
  int i = blockIdx.x * blockDim.x + threadIdx.x;   // node*8 + head
  if (i >= n * 8) return;
  int node = i >> 3, hd = i & 7;
  const float* __restrict__ hp = h + (size_t)node * 8 * C + (size_t)hd * C;
  const float* __restrict__ as = a_src + hd * C;
  const float* __restrict__ ad = a_dst + hd * C;
  float s1 = 0.f, s2 = 0.f;
  for (int c = 0; c < C; ++c) { s1 += hp[c] * as[c]; s2 += hp[c] * ad[c]; }
  ssrc[i] = s1;
  sdst[i] = s2;
}

// ---------------------------------------------------------------------------
// Edge helpers: edges 0..E-1 come from edge_index, E..E+n-1 are self loops.
// ---------------------------------------------------------------------------
__device__ __forceinline__ void edge_pair(const int* __restrict__ ei, long long E,
                                          long long idx, int& s, int& d) {
  if (idx < E) { s = ei[idx]; d = ei[E + idx]; }
  else { s = d = (int)(idx - E); }
}

__device__ __forceinline__ float leaky02(float x) {
  return x > 0.f ? x : 0.2f * x;
}

// Native CDNA5 float atomic max (GLOBAL_ATOMIC_MAX_NUM_F32).
__device__ __forceinline__ void atomic_max_f32(float* addr, float v) {
  __hip_atomic_fetch_max(addr, v, __ATOMIC_RELAXED, __HIP_MEMORY_SCOPE_AGENT);
}

// Pass 1: segment max of leaky_relu(s_src[src]+s_dst[dst]) over dst.
__global__ void edge_max(const int* __restrict__ ei, long long E, int n,
                         const float* __restrict__ ssrc,
                         const float* __restrict__ sdst,
                         float* __restrict__ m) {
  long long tot = (E + n) * 8;
  long long i = (long long)blockIdx.x * blockDim.x + threadIdx.x;
  if (i >= tot) return;
  long long eidx = i >> 3; int hd = (int)(i & 7);
  int s, d; edge_pair(ei, E, eidx, s, d);
  float e = leaky02(ssrc[(size_t)s * 8 + hd] + sdst[(size_t)d * 8 + hd]);
  atomic_max_f32(m + (size_t)d * 8 + hd, e);
}

// Pass 2: denom[dst,h] += exp(e - m[dst,h])  (recompute e).
__global__ void edge_denom(const int* __restrict__ ei, long long E, int n,
                           const float* __restrict__ ssrc,
                           const float* __restrict__ sdst,
                           const float* __restrict__ m,
                           float* __restrict__ den) {
  long long tot = (E + n) * 8;
  long long i = (long long)blockIdx.x * blockDim.x + threadIdx.x;
  if (i >= tot) return;
  long long eidx = i >> 3; int hd = (int)(i & 7);
  int s, d; edge_pair(ei, E, eidx, s, d);
  float e = leaky02(ssrc[(size_t)s * 8 + hd] + sdst[(size_t)d * 8 + hd]);
  float ex = expf(e - m[(size_t)d * 8 + hd]);
  atomicAdd(&den[(size_t)d * 8 + hd], ex);
}

// Pass 3: out[dst,h,:] += alpha * h[src,h,:]  (recompute alpha).
// Templated on C so the channel loop fully unrolls; C==8 rows are 32B-aligned
// so the h[src] gather uses b128 loads.
template <int C>
__global__ void edge_aggr(const int* __restrict__ ei, long long E, int n,
                          const float* __restrict__ ssrc,
                          const float* __restrict__ sdst,
                          const float* __restrict__ m,
                          const float* __restrict__ den,
                          const float* __restrict__ h,
                          float* __restrict__ out) {
  long long tot = (E + n) * 8;
  long long i = (long long)blockIdx.x * blockDim.x + threadIdx.x;
  if (i >= tot) return;
  long long eidx = i >> 3; int hd = (int)(i & 7);
  int s, d; edge_pair(ei, E, eidx, s, d);
  size_t di = (size_t)d * 8 + hd;
  float e = leaky02(ssrc[(size_t)s * 8 + hd] + sdst[di]);
  float alpha = expf(e - m[di]) / (den[di] + 1e-16f);
  const float* __restrict__ hp = h + (size_t)s * 8 * C + (size_t)hd * C;
  float* __restrict__ op = out + (size_t)d * 8 * C + (size_t)hd * C;

  float msg[C];
  if constexpr (C == 8) {
    const float4* __restrict__ hp4 = reinterpret_cast<const float4*>(hp);
    float4 v0 = hp4[0], v1 = hp4[1];
    msg[0] = v0.x; msg[1] = v0.y; msg[2] = v0.z; msg[3] = v0.w;
    msg[4] = v1.x; msg[5] = v1.y; msg[6] = v1.z; msg[7] = v1.w;
  } else {
#pragma unroll
    for (int c = 0; c < C; ++c) msg[c] = hp[c];
  }
#pragma unroll
  for (int c = 0; c < C; ++c) atomicAdd(&op[c], msg[c] * alpha);
}

// out = elu(in + bias), elementwise over [n, F] (in-place OK).
__global__ void bias_elu(const float* __restrict__ in,
                         const float* __restrict__ bias,
                         float* __restrict__ out, int n, int F) {
  long long i = (long long)blockIdx.x * blockDim.x + threadIdx.x;
  if (i >= (long long)n * F) return;
  float v = in[i] + bias[(int)(i % F)];
  out[i] = v > 0.f ? v : (expf(v) - 1.f);
}

// out = log_softmax(in + bias) per row of F elements.
__global__ void bias_logsoftmax(const float* __restrict__ in,
                                const float* __restrict__ bias,
                                float* __restrict__ out, int n, int F) {
  int node = blockIdx.x * blockDim.x + threadIdx.x;
  if (node >= n) return;
  const float* __restrict__ ip = in + (size_t)node * F;
  float mx = -__builtin_inff();
  for (int f = 0; f < F; ++f) mx = fmaxf(mx, ip[f] + bias[f]);
  float s = 0.f;
  for (int f = 0; f < F; ++f) s += expf(ip[f] + bias[f] - mx);
  float lse = logf(s) + mx;
  float* __restrict__ op = out + (size_t)node * F;
  for (int f = 0; f < F; ++f) op[f] = ip[f] + bias[f] - lse;
}

// ---------------------------------------------------------------------------
// Host-side launch
// ---------------------------------------------------------------------------
extern "C" void kernel_launch(void* const* d_in, const int* in_sizes, int n_in,
                              void* d_out, int out_size, void* d_ws, size_t ws_size,
                              hipStream_t stream) {
  const float* x      = (const float*)d_in[0];
  const int*   ei     = (const int*)  d_in[1];
  const float* W1     = (const float*)d_in[2];
  const float* a_src1 = (const float*)d_in[3];
  const float* a_dst1 = (const float*)d_in[4];
  const float* b1     = (const float*)d_in[5];
  const float* W2     = (const float*)d_in[6];
  const float* a_src2 = (const float*)d_in[7];
  const float* a_dst2 = (const float*)d_in[8];
  const float* b2     = (const float*)d_in[9];
  float* out = (float*)d_out;

  const int       n = in_sizes[0] / 512;       // 100000
  const long long E = in_sizes[1] / 2;         // 3200000
  const int F1 = 64, F2 = 80, C1 = 8, C2 = 10;

  // Workspace carve-out (floats)
  float* ws = (float*)d_ws;
  size_t off = 0;
  auto alloc = [&](size_t cnt) { float* p = ws + off; off += cnt; return p; };
  float* h1    = alloc((size_t)n * F1);
  float* ssrc1 = alloc((size_t)n * 8);
  float* sdst1 = alloc((size_t)n * 8);
  float* m1    = alloc((size_t)n * 8);
  float* den1  = alloc((size_t)n * 8);
  float* o1    = alloc((size_t)n * F1);
  float* h2    = alloc((size_t)n * F2);
  float* ssrc2 = alloc((size_t)n * 8);
  float* sdst2 = alloc((size_t)n * 8);
  float* m2    = alloc((size_t)n * 8);
  float* den2  = alloc((size_t)n * 8);
  float* o2    = alloc((size_t)n * F2);
  (void)ws_size; (void)n_in; (void)out_size;

  const int TB = 256;
  auto blocks = [&](long long tot) { return (unsigned)((tot + TB - 1) / TB); };
  const long long nh   = (long long)n * 8;
  const long long eh   = (E + n) * 8;
  const float NEG_INF  = -__builtin_inff();

  // --- init accumulators ---
  fill_f32<<<blocks(nh), TB, 0, stream>>>(m1,   NEG_INF, nh);
  fill_f32<<<blocks(nh), TB, 0, stream>>>(den1, 0.f,     nh);
  fill_f32<<<blocks((long long)n*F1), TB, 0, stream>>>(o1, 0.f, (long long)n*F1);
  fill_f32<<<blocks(nh), TB, 0, stream>>>(m2,   NEG_INF, nh);
  fill_f32<<<blocks(nh), TB, 0, stream>>>(den2, 0.f,     nh);
  fill_f32<<<blocks((long long)n*F2), TB, 0, stream>>>(o2, 0.f, (long long)n*F2);

  // --- layer 1: h1 = x @ W1 (WMMA, A panel staged in LDS) ---
  {
    const int K = 512;
    size_t shmem = (size_t)16 * (K + 4) * sizeof(float);
    wmma_gemm_f32<<<dim3(n / 16), 32 * (F1 / 16), shmem, stream>>>(x, W1, h1, n, K, F1);
  }
  attn_scores<<<blocks(nh), TB, 0, stream>>>(h1, a_src1, a_dst1, ssrc1, sdst1, n, C1);
  edge_max  <<<blocks(eh), TB, 0, stream>>>(ei, E, n, ssrc1, sdst1, m1);
  edge_denom<<<blocks(eh), TB, 0, stream>>>(ei, E, n, ssrc1, sdst1, m1, den1);
  edge_aggr<C1><<<blocks(eh), TB, 0, stream>>>(ei, E, n, ssrc1, sdst1, m1, den1, h1, o1);
  bias_elu  <<<blocks((long long)n*F1), TB, 0, stream>>>(o1, b1, o1, n, F1);

  // --- layer 2: h2 = elu_out @ W2 ---
  {
    const int K = F1;
    size_t shmem = (size_t)16 * (K + 4) * sizeof(float);
    wmma_gemm_f32<<<dim3(n / 16), 32 * (F2 / 16), shmem, stream>>>(o1, W2, h2, n, K, F2);
  }
  attn_scores<<<blocks(nh), TB, 0, stream>>>(h2, a_src2, a_dst2, ssrc2, sdst2, n, C2);
  edge_max  <<<blocks(eh), TB, 0, stream>>>(ei, E, n, ssrc2, sdst2, m2);
  edge_denom<<<blocks(eh), TB, 0, stream>>>(ei, E, n, ssrc2, sdst2, m2, den2);
  edge_aggr<C2><<<blocks(eh), TB, 0, stream>>>(ei, E, n, ssrc2, sdst2, m2, den2, h2, o2);

  // --- output: bias + log_softmax ---
  bias_logsoftmax<<<blocks(n), TB, 0, stream>>>(o2, b2, out, n, F2);
}